// DCRNNModel_24610162606124
// MI455X (gfx1250) — compile-verified
//
#include <hip/hip_runtime.h>
#include <hip/hip_bf16.h>

typedef __attribute__((ext_vector_type(2))) float v2f;
typedef __attribute__((ext_vector_type(8))) float v8f;

#define N_NODES 100000
#define F_IN    16
#define HIDN    32
#define OUTD    12
#define CINE    48            // effective input channels: [x | T_o | T_i]
#define TILES   (N_NODES/16)  // 6250, exact

// fp32 WMMA, D = A(16x4) * B(4x16) + C(16x16)
__device__ __forceinline__ v8f wmma_k4(v2f a, v2f b, v8f c) {
    return __builtin_amdgcn_wmma_f32_16x16x4_f32(false, a, false, b, (short)0, c,
                                                 false, false);
}

// ---- weighted degree accumulation over edges -------------------------------
__global__ void deg_kernel(const int* __restrict__ src, const int* __restrict__ dst,
                           const float* __restrict__ w,
                           float* __restrict__ deg_out, float* __restrict__ deg_in, int E) {
    int e = blockIdx.x * blockDim.x + threadIdx.x;
    if (e < E) {
        float we = w[e];
        unsafeAtomicAdd(&deg_out[src[e]], we);
        unsafeAtomicAdd(&deg_in[dst[e]],  we);
    }
}

// in-place reciprocal (matches reference 1/deg, inf on zero degree)
__global__ void inv_kernel(float* __restrict__ p, int n) {
    int i = blockIdx.x * blockDim.x + threadIdx.x;
    if (i < n) p[i] = 1.0f / p[i];
}

// ---- diffusion scatter: one thread per (edge, feature) ---------------------
__global__ void scatter_kernel(const int* __restrict__ src, const int* __restrict__ dst,
                               const float* __restrict__ x,
                               const float* __restrict__ inv_out, const float* __restrict__ inv_in,
                               float* __restrict__ T_o, float* __restrict__ T_i, int E) {
    long tid = (long)blockIdx.x * blockDim.x + threadIdx.x;
    if (tid < (long)E * 16) {
        int e = (int)(tid >> 4);
        int f = (int)(tid & 15);
        int s = src[e], d = dst[e];
        unsafeAtomicAdd(&T_o[d * 16 + f], inv_out[s] * x[s * 16 + f]);
        unsafeAtomicAdd(&T_i[s * 16 + f], inv_in[d]  * x[d * 16 + f]);
    }
}

// ---- fold weights into effective stacked matrices --------------------------
// W shape (2, K=2, CIN=48, HID=32) row-major; only first 16 input rows matter (H0==0).
// B_g rows 0-15 : W[0,0][0:16] + W[1,0][0:16]  (offsets 0 and 3072)
// B_g rows 16-31: W[0,1][0:16]                 (offset 1536)   (applied to T_o)
// B_g rows 32-47: W[1,1][0:16]                 (offset 4608)   (applied to T_i)
__global__ void prep_kernel(const float* __restrict__ Wz, const float* __restrict__ Wh,
                            const float* __restrict__ linW, const float* __restrict__ linb,
                            float* __restrict__ Bz, float* __restrict__ Bh,
                            float* __restrict__ LW, float* __restrict__ LB) {
    int t = blockIdx.x * blockDim.x + threadIdx.x;
    if (t < CINE * HIDN) {
        int r = t >> 5, c = t & 31;
        float vz, vh;
        if (r < 16) {
            vz = Wz[r * 32 + c] + Wz[3072 + r * 32 + c];
            vh = Wh[r * 32 + c] + Wh[3072 + r * 32 + c];
        } else if (r < 32) {
            int i = r - 16;
            vz = Wz[1536 + i * 32 + c];
            vh = Wh[1536 + i * 32 + c];
        } else {
            int i = r - 32;
            vz = Wz[4608 + i * 32 + c];
            vh = Wh[4608 + i * 32 + c];
        }
        Bz[t] = vz; Bh[t] = vh;
    }
    if (t < HIDN * 16) {   // lin_W [32,12] padded to [32,16]
        int k = t >> 4, o = t & 15;
        LW[t] = (o < OUTD) ? linW[k * OUTD + o] : 0.0f;
    }
    if (t < 16) LB[t] = (t < OUTD) ? linb[t] : 0.0f;
}

// ---- fused gate GEMMs + activations + output GEMM (fp32 WMMA) --------------
// 2 waves per block, one 16-node tile per wave; 6250 tiles = 3125 blocks exactly,
// so EXEC is always all-1s for every WMMA.
__global__ __launch_bounds__(64)
void gates_kernel(const float* __restrict__ x,
                  const float* __restrict__ T_o, const float* __restrict__ T_i,
                  const float* __restrict__ Bz, const float* __restrict__ Bh,
                  const float* __restrict__ bz, const float* __restrict__ bh,
                  const float* __restrict__ LW, const float* __restrict__ LB,
                  float* __restrict__ out) {
    __shared__ float sBz[CINE * HIDN];
    __shared__ float sBh[CINE * HIDN];
    __shared__ float sLW[HIDN * 16];
    __shared__ float sLB[16];
    __shared__ float sH[2][16 * 34];   // per-wave H tile, padded stride

    int tid = threadIdx.x;
    for (int i = tid; i < CINE * HIDN; i += 64) { sBz[i] = Bz[i]; sBh[i] = Bh[i]; }
    for (int i = tid; i < HIDN * 16; i += 64) sLW[i] = LW[i];
    if (tid < 16) sLB[tid] = LB[tid];
    __syncthreads();

    int wave  = tid >> 5;
    int lane  = tid & 31;
    int tile  = blockIdx.x * 2 + wave;
    int m     = lane & 15;       // row within tile (A row, B/C col)
    int half  = lane >> 4;       // 0/1
    int khalf = half * 2;        // K sub-offset for 32-bit A/B fragments
    int n     = m;               // output column within 16-wide tile

    // ---- A fragments: virtual row [x | T_o | T_i] of the lane's node ----
    int node = tile * 16 + m;
    const float* xr  = x   + node * F_IN;
    const float* tor = T_o + node * F_IN;
    const float* tir = T_i + node * F_IN;
    v2f a[12];
#pragma unroll
    for (int t = 0; t < 4; ++t) a[t]     = *(const v2f*)(xr  + 4 * t + khalf);
#pragma unroll
    for (int t = 0; t < 4; ++t) a[4 + t] = *(const v2f*)(tor + 4 * t + khalf);
#pragma unroll
    for (int t = 0; t < 4; ++t) a[8 + t] = *(const v2f*)(tir + 4 * t + khalf);

    // accumulators preloaded with the gate biases (broadcast along rows)
    float bz0 = bz[n], bz1 = bz[16 + n];
    float bh0 = bh[n], bh1 = bh[16 + n];
    v8f az0, az1, ah0, ah1;
#pragma unroll
    for (int r = 0; r < 8; ++r) { az0[r] = bz0; az1[r] = bz1; ah0[r] = bh0; ah1[r] = bh1; }

#pragma unroll
    for (int t = 0; t < 12; ++t) {
        int kb = 4 * t + khalf;
        v2f b0, b1;
        b0.x = sBz[kb * HIDN + n];        b0.y = sBz[(kb + 1) * HIDN + n];
        b1.x = sBz[kb * HIDN + 16 + n];   b1.y = sBz[(kb + 1) * HIDN + 16 + n];
        az0 = wmma_k4(a[t], b0, az0);
        az1 = wmma_k4(a[t], b1, az1);
        b0.x = sBh[kb * HIDN + n];        b0.y = sBh[(kb + 1) * HIDN + n];
        b1.x = sBh[kb * HIDN + 16 + n];   b1.y = sBh[(kb + 1) * HIDN + 16 + n];
        ah0 = wmma_k4(a[t], b0, ah0);
        ah1 = wmma_k4(a[t], b1, ah1);
    }

    // activations: H = (1-Z)*tanh(Gh), then relu; stage to LDS in row-major
    float* hrow = &sH[wave][0];
#pragma unroll
    for (int r = 0; r < 8; ++r) {
        int row = r + 8 * half;
        float z0 = 1.0f / (1.0f + __expf(-az0[r]));
        float v0 = fmaxf((1.0f - z0) * tanhf(ah0[r]), 0.0f);
        hrow[row * 34 + n] = v0;
        float z1 = 1.0f / (1.0f + __expf(-az1[r]));
        float v1 = fmaxf((1.0f - z1) * tanhf(ah1[r]), 0.0f);
        hrow[row * 34 + 16 + n] = v1;
    }
    __syncthreads();

    // ---- output GEMM: relu(H)[16,32] @ lin_W_pad[32,16] + lin_b ----
    v8f ao;
    float lb = sLB[n];
#pragma unroll
    for (int r = 0; r < 8; ++r) ao[r] = lb;
#pragma unroll
    for (int t = 0; t < 8; ++t) {
        int kb = 4 * t + khalf;
        v2f av, bv;
        av.x = hrow[m * 34 + kb];     av.y = hrow[m * 34 + kb + 1];
        bv.x = sLW[kb * 16 + n];      bv.y = sLW[(kb + 1) * 16 + n];
        ao = wmma_k4(av, bv, ao);
    }
    if (n < OUTD) {
#pragma unroll
        for (int r = 0; r < 8; ++r) {
            int row = tile * 16 + r + 8 * half;
            out[row * OUTD + n] = ao[r];
        }
    }
}

extern "C" void kernel_launch(void* const* d_in, const int* in_sizes, int n_in,
                              void* d_out, int out_size, void* d_ws, size_t ws_size,
                              hipStream_t stream) {
    const float* x    = (const float*)d_in[0];
    const int*   ei   = (const int*)d_in[1];
    const float* ew   = (const float*)d_in[2];
    const float* Wz   = (const float*)d_in[3];
    const float* bz   = (const float*)d_in[4];
    // d_in[5] = W_r, d_in[6] = b_r: mathematically unused (H0 == 0 kills the reset gate)
    const float* Wh   = (const float*)d_in[7];
    const float* bh   = (const float*)d_in[8];
    const float* linW = (const float*)d_in[9];
    const float* linb = (const float*)d_in[10];
    float* out = (float*)d_out;
    float* ws  = (float*)d_ws;

    const int N = N_NODES;
    const int E = in_sizes[2];
    const int* src = ei;
    const int* dst = ei + E;

    float* deg_out = ws;                 // N  (becomes 1/deg_out)
    float* deg_in  = ws + N;             // N  (becomes 1/deg_in)
    float* T_o     = ws + 2 * N;         // 16N
    float* T_i     = ws + 18 * N;        // 16N
    float* Bz_eff  = ws + 34 * N;        // 1536
    float* Bh_eff  = Bz_eff + CINE * HIDN;
    float* LW      = Bh_eff + CINE * HIDN;   // 512
    float* LB      = LW + HIDN * 16;         // 16

    // zero accumulation buffers (graph-capturable)
    hipMemsetAsync(ws, 0, (size_t)34 * N * sizeof(float), stream);

    prep_kernel<<<6, 256, 0, stream>>>(Wz, Wh, linW, linb, Bz_eff, Bh_eff, LW, LB);

    deg_kernel<<<(E + 255) / 256, 256, 0, stream>>>(src, dst, ew, deg_out, deg_in, E);
    inv_kernel<<<(2 * N + 255) / 256, 256, 0, stream>>>(ws, 2 * N);

    long stotal = (long)E * 16;
    scatter_kernel<<<(int)((stotal + 255) / 256), 256, 0, stream>>>(
        src, dst, x, deg_out, deg_in, T_o, T_i, E);

    gates_kernel<<<TILES / 2, 64, 0, stream>>>(x, T_o, T_i, Bz_eff, Bh_eff,
                                               bz, bh, LW, LB, out);
}